// RelPos_1279900254323
// MI455X (gfx1250) — compile-verified
//
#include <hip/hip_runtime.h>
#include <stdint.h>

// RelPos: out[p*128 + c] = W.T[clamp(r_i - r_j, -32, 32) + 32][c] + b[c]
// Pure gather + bias broadcast: 302 MB of output writes, ~13 us at 23.3 TB/s.
// Strategy:
//   1. Async-copy bias + residue_index into LDS (gfx1250 ASYNCcnt path).
//   2. Build table_pb[65][128] = W.T + b in LDS using v_wmma_f32_16x16x4_f32
//      (rank-1 broadcast add: A = ones in K=0 column, B = bias in K=0 row,
//      C = W.T tile  =>  D = bias + W.T, exact f32, same as reference).
//   3. Stream: one wave per (i,j) pair, lane l stores float4 of channels
//      [4l..4l+3] with a non-temporal b128 store (output > L2, don't thrash).

typedef __attribute__((ext_vector_type(2))) float v2f;
typedef __attribute__((ext_vector_type(4))) float v4f;
typedef __attribute__((ext_vector_type(8))) float v8f;
typedef __attribute__((ext_vector_type(4))) int v4i;

#define NRES 768
#define CZ 128
#define NBINS 65
#define MAXREL 32

// LDS layout (float offsets)
#define TAB_F 0                  // table_pb[65][128]  (33280 B)
#define BST_F (NBINS * CZ)       // bias[128]          (512 B)
#define RST_F (BST_F + CZ)       // residue_index[768] (3072 B, as int)
#define SMEM_F (RST_F + NRES)    // 9216 floats = 36864 B total

__device__ __forceinline__ void async_copy_b128(const void* g, void* l) {
#if __has_builtin(__builtin_amdgcn_global_load_async_to_lds_b128)
  __builtin_amdgcn_global_load_async_to_lds_b128(
      (__attribute__((address_space(1))) v4i*)g,
      (__attribute__((address_space(3))) v4i*)l, 0, 0);
#else
  uint32_t loff = (uint32_t)(uintptr_t)(__attribute__((address_space(3))) void*)l;
  uint64_t ga = (uint64_t)(uintptr_t)g;
  asm volatile("global_load_async_to_lds_b128 %0, %1, off"
               :: "v"(loff), "v"(ga) : "memory");
#endif
}

__device__ __forceinline__ void wait_async_zero() {
#if __has_builtin(__builtin_amdgcn_s_wait_asynccnt)
  __builtin_amdgcn_s_wait_asynccnt(0);
#else
  asm volatile("s_wait_asynccnt 0" ::: "memory");
#endif
}

__global__ __launch_bounds__(256) void relpos_kernel(
    const int* __restrict__ resi, const float* __restrict__ W,
    const float* __restrict__ bias, float* __restrict__ out) {
  __shared__ float smem[SMEM_F];
  const int tid = (int)threadIdx.x;
  const int lane = tid & 31;
  const int wave = tid >> 5;

  // ---- Phase 1: async-stage bias (32 x b128) + residues (192 x b128) to LDS.
  if (tid < 224) {
    const char* g;
    int loff;
    if (tid < 32) {
      g = (const char*)bias + tid * 16;
      loff = BST_F * 4 + tid * 16;
    } else {
      g = (const char*)resi + (tid - 32) * 16;
      loff = RST_F * 4 + (tid - 32) * 16;
    }
    async_copy_b128(g, (char*)smem + loff);
  }
  wait_async_zero();
  __syncthreads();

  // ---- Phase 2: table_pb[k][c] = W[c][k] + bias[c] via WMMA rank-1 update.
#if __has_builtin(__builtin_amdgcn_wmma_f32_16x16x4_f32)
  {
    const float* bs = smem + BST_F;
    const int l16 = lane & 15;
    const int moff = (lane >> 4) * 8;  // C/D rows: lanes 16-31 hold M+8
    // A (16x4): A[m][k] = (k==0) ? 1 : 0. K=0 lives in VGPR0 lanes 0-15.
    v2f amat;
    amat.x = (lane < 16) ? 1.0f : 0.0f;
    amat.y = 0.0f;
    for (int t = wave; t < 32; t += 8) {        // full wave per tile, EXEC all-1s
      const int k0 = (t >> 3) * 16;             // 0,16,32,48
      const int c0 = (t & 7) * 16;              // 0..112
      // B (4x16): B[k][n] = (k==0) ? bias[c0+n] : 0.
      v2f bmat;
      bmat.x = (lane < 16) ? bs[c0 + l16] : 0.0f;
      bmat.y = 0.0f;
      // C (16x16): C[m][n] = W[c0+n][k0+m]  (transposed gather from W[128][65])
      v8f cmat;
#pragma unroll
      for (int r = 0; r < 8; ++r)
        cmat[r] = W[(c0 + l16) * NBINS + k0 + moff + r];
      v8f acc = __builtin_amdgcn_wmma_f32_16x16x4_f32(
          false, amat, false, bmat, (short)0, cmat, false, false);
      // D[m][n] = bias[c0+n] + W.T[k0+m][c0+n]  ->  table_pb[k0+m][c0+n]
#pragma unroll
      for (int r = 0; r < 8; ++r)
        smem[(k0 + moff + r) * CZ + c0 + l16] = acc[r];
    }
    // leftover bin 64
    if (tid < CZ) smem[64 * CZ + tid] = W[tid * NBINS + 64] + bs[tid];
  }
#else
  for (int idx = tid; idx < NBINS * CZ; idx += 256) {
    int k = idx >> 7, cc = idx & 127;
    smem[idx] = W[cc * NBINS + k] + smem[BST_F + cc];
  }
#endif
  __syncthreads();

  // ---- Phase 3: stream rows. One wave per (i,j); lane l -> channels 4l..4l+3.
  const int* rres = (const int*)(smem + RST_F);
  const uint32_t total = (uint32_t)NRES * (uint32_t)NRES;
  const uint32_t stride = gridDim.x * 8u;
  for (uint32_t p = blockIdx.x * 8u + (uint32_t)wave; p < total; p += stride) {
    uint32_t i = p / (uint32_t)NRES;
    uint32_t j = p - i * (uint32_t)NRES;
    int dd = rres[i] - rres[j];
    dd = dd < -MAXREL ? -MAXREL : (dd > MAXREL ? MAXREL : dd);
    dd += MAXREL;
    v4f v = *(const v4f*)(smem + dd * CZ + lane * 4);
    __builtin_nontemporal_store(v, (v4f*)(out + (size_t)p * CZ + (uint32_t)(lane * 4)));
  }
}

extern "C" void kernel_launch(void* const* d_in, const int* in_sizes, int n_in,
                              void* d_out, int out_size, void* d_ws, size_t ws_size,
                              hipStream_t stream) {
  (void)in_sizes; (void)n_in; (void)out_size; (void)d_ws; (void)ws_size;
  const int* resi = (const int*)d_in[0];     // residue_index [1,768] int32
  const float* W = (const float*)d_in[1];    // W [128][65] f32 (row-major)
  const float* b = (const float*)d_in[2];    // b [128] f32
  float* out = (float*)d_out;                // [1,768,768,128] f32
  dim3 grid(2048), block(256);
  hipLaunchKernelGGL(relpos_kernel, grid, block, 0, stream, resi, W, b, out);
}